// GICPModel_73882027425857
// MI455X (gfx1250) — compile-verified
//
#include <hip/hip_runtime.h>

typedef float v4f __attribute__((ext_vector_type(4)));
typedef float v2f __attribute__((ext_vector_type(2)));
typedef float v8f __attribute__((ext_vector_type(8)));

#define BLOCK 256
#define NBLK  1024

__device__ __forceinline__ float dot4(v4f a, float r0, float r1, float r2, float r3) {
    return a.x * r0 + a.y * r1 + a.z * r2 + a.w * r3;
}

// Kernel 1: grid-stride streaming pass; per-block partial sums -> d_ws
__global__ __launch_bounds__(BLOCK) void gicp_partial(
    const float* __restrict__ tmat,   // [4,4]
    const float* __restrict__ srcp,   // [N,4]
    const float* __restrict__ tarp,   // [N,4]
    const float* __restrict__ csrc,   // [N,4,4]
    const float* __restrict__ ctar,   // [N,4,4]
    float* __restrict__ partial, int n)
{
    // Transform rows (only first 3 rows of T are needed): kept in scalars -> SGPR/VGPR
    const float R00 = tmat[0],  R01 = tmat[1],  R02 = tmat[2],  R03 = tmat[3];
    const float R10 = tmat[4],  R11 = tmat[5],  R12 = tmat[6],  R13 = tmat[7];
    const float R20 = tmat[8],  R21 = tmat[9],  R22 = tmat[10], R23 = tmat[11];

    const int tid    = blockIdx.x * BLOCK + threadIdx.x;
    const int stride = gridDim.x * BLOCK;

    float acc = 0.0f;

    for (int i = tid; i < n; i += stride) {
        // Speculative prefetch of next iteration's covariance lines (global_prefetch_b8)
        __builtin_prefetch(csrc + (size_t)(i + stride) * 16, 0, 0);
        __builtin_prefetch(ctar + (size_t)(i + stride) * 16, 0, 0);

        const v4f* ps = (const v4f*)(csrc + (size_t)i * 16);
        const v4f* pt = (const v4f*)(ctar + (size_t)i * 16);

        // Non-temporal B128 loads: single-pass stream, don't pollute L2
        v4f cs0 = __builtin_nontemporal_load(ps + 0);
        v4f cs1 = __builtin_nontemporal_load(ps + 1);
        v4f cs2 = __builtin_nontemporal_load(ps + 2);
        v4f cs3 = __builtin_nontemporal_load(ps + 3);
        v4f ct0 = __builtin_nontemporal_load(pt + 0);
        v4f ct1 = __builtin_nontemporal_load(pt + 1);
        v4f ct2 = __builtin_nontemporal_load(pt + 2);
        v4f sp  = __builtin_nontemporal_load((const v4f*)(srcp + (size_t)i * 4));
        v4f tp  = __builtin_nontemporal_load((const v4f*)(tarp + (size_t)i * 4));

        const float r0 = tp.x - sp.x;
        const float r1 = tp.y - sp.y;
        const float r2 = tp.z - sp.z;

        // M = R(3x4) @ C_src(4x4): row i of M = sum_k R[i][k] * row_k(C)
        v4f m0 = R00 * cs0 + R01 * cs1 + R02 * cs2 + R03 * cs3;
        v4f m1 = R10 * cs0 + R11 * cs1 + R12 * cs2 + R13 * cs3;
        v4f m2 = R20 * cs0 + R21 * cs1 + R22 * cs2 + R23 * cs3;

        // S = C_tar[:3,:3] + M @ R^T
        const float s00 = ct0.x + dot4(m0, R00, R01, R02, R03);
        const float s01 = ct0.y + dot4(m0, R10, R11, R12, R13);
        const float s02 = ct0.z + dot4(m0, R20, R21, R22, R23);
        const float s10 = ct1.x + dot4(m1, R00, R01, R02, R03);
        const float s11 = ct1.y + dot4(m1, R10, R11, R12, R13);
        const float s12 = ct1.z + dot4(m1, R20, R21, R22, R23);
        const float s20 = ct2.x + dot4(m2, R00, R01, R02, R03);
        const float s21 = ct2.y + dot4(m2, R10, R11, R12, R13);
        const float s22 = ct2.z + dot4(m2, R20, R21, R22, R23);

        // q = r^T S^{-1} r = (r^T cof(S) r) / det(S)   (no explicit inverse)
        const float c00 =  (s11 * s22 - s12 * s21);
        const float c01 = -(s10 * s22 - s12 * s20);
        const float c02 =  (s10 * s21 - s11 * s20);
        const float c10 = -(s01 * s22 - s02 * s21);
        const float c11 =  (s00 * s22 - s02 * s20);
        const float c12 = -(s00 * s21 - s01 * s20);
        const float c20 =  (s01 * s12 - s02 * s11);
        const float c21 = -(s00 * s12 - s02 * s10);
        const float c22 =  (s00 * s11 - s01 * s10);

        const float det = s00 * c00 + s01 * c01 + s02 * c02;

        const float num = r0 * (c00 * r0 + c01 * r1 + c02 * r2)
                        + r1 * (c10 * r0 + c11 * r1 + c12 * r2)
                        + r2 * (c20 * r0 + c21 * r1 + c22 * r2);

        acc += num / det;
    }

    // ---- Wave reduction via V_WMMA_F32_16X16X4_F32 ----
    // A(16x4): lane m holds K0,K1 = acc_m; lane m+16 holds K2,K3 = acc_{m+16}
    //   -> row m of A = (acc_m, acc_m, acc_{m+16}, acc_{m+16})
    // B = all 0.5 (layout-invariant): D(m,n) = acc_m + acc_{m+16}
    v2f a = {acc, acc};
    v2f b = {0.5f, 0.5f};
    v8f c = {};
    v8f d = __builtin_amdgcn_wmma_f32_16x16x4_f32(
        /*neg_a=*/false, a, /*neg_b=*/false, b,
        /*c_mod=*/(short)0, c, /*reuse_a=*/false, /*reuse_b=*/false);

    float t = d[0] + d[1] + d[2] + d[3] + d[4] + d[5] + d[6] + d[7];
    // lanes 0-15 hold sum over M=0..7; lanes 16-31 hold sum over M=8..15
    t += __shfl_xor(t, 16, 32);  // full wave32 total in every lane

    __shared__ float wsum[BLOCK / 32];
    const int lane = threadIdx.x & 31;
    const int wid  = threadIdx.x >> 5;
    if (lane == 0) wsum[wid] = t;
    __syncthreads();
    if (threadIdx.x == 0) {
        float s = 0.0f;
        #pragma unroll
        for (int w = 0; w < BLOCK / 32; ++w) s += wsum[w];
        partial[blockIdx.x] = s;   // deterministic per-block partial
    }
}

// Kernel 2: single block, fixed-order reduction of partials -> scalar output
__global__ __launch_bounds__(256) void gicp_final(
    const float* __restrict__ partial, int nblk, float scale,
    float* __restrict__ out)
{
    __shared__ float sm[256];
    float s = 0.0f;
    for (int i = threadIdx.x; i < nblk; i += 256) s += partial[i];
    sm[threadIdx.x] = s;
    __syncthreads();
    for (int off = 128; off > 0; off >>= 1) {
        if (threadIdx.x < off) sm[threadIdx.x] += sm[threadIdx.x + off];
        __syncthreads();
    }
    if (threadIdx.x == 0) out[0] = sm[0] * scale;
}

extern "C" void kernel_launch(void* const* d_in, const int* in_sizes, int n_in,
                              void* d_out, int out_size, void* d_ws, size_t ws_size,
                              hipStream_t stream) {
    const float* tmat = (const float*)d_in[0];
    const float* srcp = (const float*)d_in[1];
    const float* tarp = (const float*)d_in[2];
    const float* csrc = (const float*)d_in[3];
    const float* ctar = (const float*)d_in[4];

    const int n = in_sizes[1] / 4;          // N points
    float* partial = (float*)d_ws;          // NBLK floats of scratch
    float* out = (float*)d_out;

    gicp_partial<<<NBLK, BLOCK, 0, stream>>>(tmat, srcp, tarp, csrc, ctar, partial, n);
    gicp_final<<<1, 256, 0, stream>>>(partial, NBLK, 0.5f / (float)n, out);
}